// SimpleAttention_4999341933116
// MI455X (gfx1250) — compile-verified
//
#include <hip/hip_runtime.h>

// SimpleAttention on MI455X (gfx1250, wave32).
//
// Reference:  q = x@Wq^T+bq (and k, v);  attn = outer(q,k) per batch;
//             out[b,i] = softmax_j(q_i*k_j) . v  — rank-1 logits, so the
//             [B,D,D] tensor is never materialized.
//
// Stage 1: three 2048x512x512 f32 GEMMs via V_WMMA_F32_16X16X4_F32.
// Stage 2: per-row streaming softmax-weighted sum. TRANS(exp)-bound;
//          k/v interleaved as float2 in LDS so the hot loop does ONE
//          broadcast ds_load_b64 per j instead of two b32 loads.

typedef __attribute__((ext_vector_type(2))) float v2f;
typedef __attribute__((ext_vector_type(8))) float v8f;

#define BB 2048
#define DD 512

// ---------------------------------------------------------------------------
// Stage 1: Y = X @ W^T + bias for proj in {q,k,v}.
// One wave computes a 16(M) x 64(N) strip: 4 f32 accumulators, K-loop of 128
// steps of K=4. A-fragment (activations) is loaded once per step and reused
// across 4 WMMAs (weights strip).
// ---------------------------------------------------------------------------
__global__ __launch_bounds__(256) void simpleattn_proj_gemm(
    const float* __restrict__ query, const float* __restrict__ key_,
    const float* __restrict__ value,
    const float* __restrict__ Wq, const float* __restrict__ bq,
    const float* __restrict__ Wk, const float* __restrict__ bk,
    const float* __restrict__ Wv, const float* __restrict__ bv,
    float* __restrict__ qbuf, float* __restrict__ kbuf,
    float* __restrict__ vbuf)
{
    // Global wave id -> job = proj*1024 + mtile*8 + nstrip
    // (128 M-tiles of 16 rows, 8 N-strips of 64 cols, 3 projections).
    const int wave   = (blockIdx.x << 3) | (threadIdx.x >> 5);
    const int proj   = wave >> 10;
    const int rem    = wave & 1023;
    const int m0     = (rem >> 3) << 4;   // M-tile origin
    const int n0     = (rem & 7) << 6;    // N-strip origin (64 wide)

    const float* X; const float* W; const float* bias; float* Y;
    if (proj == 0)      { X = query; W = Wq; bias = bq; Y = qbuf; }
    else if (proj == 1) { X = key_;  W = Wk; bias = bk; Y = kbuf; }
    else                { X = value; W = Wv; bias = bv; Y = vbuf; }

    const int lane   = threadIdx.x & 31;
    const int laneLo = lane & 15;   // M (for A) / N (for B, C/D)
    const int laneHi = lane >> 4;   // selects K pair {0,1} vs {2,3}; M+8 in C/D

    // A fragment: A[m,k] = X[m0+m, k0+k]; lane holds row m0+laneLo,
    // K-pair 2*laneHi in 2 consecutive floats -> one v2f load per step.
    const float* aptr = X + (m0 + laneLo) * DD + (laneHi << 1);
    // B fragment: B[k,n] = W[n0+n, k0+k] (B = W^T tile); same per-lane shape
    // over W rows. Four sub-tiles at n0 + {0,16,32,48}.
    const float* bptr = W + (n0 + laneLo) * DD + (laneHi << 1);

    v8f acc0 = {}, acc1 = {}, acc2 = {}, acc3 = {};

    #pragma unroll 4
    for (int k0 = 0; k0 < DD; k0 += 4) {
        const v2f a  = *(const v2f*)(aptr + k0);
        const v2f b0 = *(const v2f*)(bptr + k0);
        const v2f b1 = *(const v2f*)(bptr + 16 * DD + k0);
        const v2f b2 = *(const v2f*)(bptr + 32 * DD + k0);
        const v2f b3 = *(const v2f*)(bptr + 48 * DD + k0);
        // 8 args: (neg_a, A, neg_b, B, c_mod, C, reuse_a, reuse_b)
        acc0 = __builtin_amdgcn_wmma_f32_16x16x4_f32(false, a, false, b0,
                                                     (short)0, acc0, false, false);
        acc1 = __builtin_amdgcn_wmma_f32_16x16x4_f32(false, a, false, b1,
                                                     (short)0, acc1, false, false);
        acc2 = __builtin_amdgcn_wmma_f32_16x16x4_f32(false, a, false, b2,
                                                     (short)0, acc2, false, false);
        acc3 = __builtin_amdgcn_wmma_f32_16x16x4_f32(false, a, false, b3,
                                                     (short)0, acc3, false, false);
    }

    // Bias depends only on N = n0 + sub*16 + laneLo: one scalar per lane
    // per sub-tile, added to all 8 accumulator VGPRs.
    const float bia0 = bias[n0 +  0 + laneLo];
    const float bia1 = bias[n0 + 16 + laneLo];
    const float bia2 = bias[n0 + 32 + laneLo];
    const float bia3 = bias[n0 + 48 + laneLo];

    // C/D layout: VGPR r, lane l -> (M = m0 + r + 8*laneHi, N = n0 + laneLo).
    float* yrow = Y + n0 + laneLo;
    #pragma unroll
    for (int r = 0; r < 8; ++r) {
        const int m = m0 + r + (laneHi << 3);
        yrow[m * DD +  0] = acc0[r] + bia0;
        yrow[m * DD + 16] = acc1[r] + bia1;
        yrow[m * DD + 32] = acc2[r] + bia2;
        yrow[m * DD + 48] = acc3[r] + bia3;
    }
}

// ---------------------------------------------------------------------------
// Stage 2: out[b,i] = sum_j exp(q_i*k_j - m_i) * v_j / sum_j exp(q_i*k_j - m_i)
// with m_i = q_i >= 0 ? q_i*kmax : q_i*kmin (exact row max of rank-1 logits).
// One block per batch row; k/v interleaved in LDS as float2 (one broadcast
// ds_load_b64 per j); 512 threads, one output element each; 512 exps per
// thread on the TRANS pipe.
// ---------------------------------------------------------------------------
__global__ __launch_bounds__(512) void simpleattn_softmax_av(
    const float* __restrict__ qbuf, const float* __restrict__ kbuf,
    const float* __restrict__ vbuf, float* __restrict__ out)
{
    __shared__ float kv[DD * 2];   // interleaved {k[j], v[j]}
    __shared__ float rmax[DD];
    __shared__ float rmin[DD];

    const int b   = blockIdx.x;
    const int tid = threadIdx.x;

    const float kj = kbuf[b * DD + tid];
    kv[2 * tid + 0] = kj;
    kv[2 * tid + 1] = vbuf[b * DD + tid];
    rmax[tid] = kj;
    rmin[tid] = kj;
    __syncthreads();

    // Block tree reduction for kmax / kmin.
    for (int s2 = DD >> 1; s2 > 0; s2 >>= 1) {
        if (tid < s2) {
            rmax[tid] = fmaxf(rmax[tid], rmax[tid + s2]);
            rmin[tid] = fminf(rmin[tid], rmin[tid + s2]);
        }
        __syncthreads();
    }
    const float kmax = rmax[0];
    const float kmin = rmin[0];

    const float s = qbuf[b * DD + tid];
    const float m = (s >= 0.0f) ? s * kmax : s * kmin;

    float num = 0.0f, den = 0.0f;
    #pragma unroll 8
    for (int j = 0; j < DD; ++j) {
        const v2f kvj = *(const v2f*)(kv + 2 * j);   // broadcast b64 load
        const float e = __expf(fmaf(s, kvj.x, -m));
        num = fmaf(e, kvj.y, num);
        den += e;
    }
    out[b * DD + tid] = num / den;
}

// ---------------------------------------------------------------------------
extern "C" void kernel_launch(void* const* d_in, const int* in_sizes, int n_in,
                              void* d_out, int out_size, void* d_ws,
                              size_t ws_size, hipStream_t stream)
{
    const float* query = (const float*)d_in[0];
    const float* key_  = (const float*)d_in[1];
    const float* value = (const float*)d_in[2];
    const float* Wq    = (const float*)d_in[3];
    const float* bq    = (const float*)d_in[4];
    const float* Wk    = (const float*)d_in[5];
    const float* bk    = (const float*)d_in[6];
    const float* Wv    = (const float*)d_in[7];
    const float* bv    = (const float*)d_in[8];
    float* out = (float*)d_out;

    float* qbuf = (float*)d_ws;          // [B, D]
    float* kbuf = qbuf + (size_t)BB * DD;
    float* vbuf = kbuf + (size_t)BB * DD;

    // 3 proj * 128 M-tiles * 8 N-strips = 3072 wave-jobs; 8 waves/block.
    simpleattn_proj_gemm<<<384, 256, 0, stream>>>(
        query, key_, value, Wq, bq, Wk, bk, Wv, bv, qbuf, kbuf, vbuf);

    simpleattn_softmax_av<<<BB, DD, 0, stream>>>(qbuf, kbuf, vbuf, out);
}